// DGLGCNEncoder_48266842472902
// MI455X (gfx1250) — compile-verified
//
#include <hip/hip_runtime.h>

typedef float v2f __attribute__((ext_vector_type(2)));
typedef float v8f __attribute__((ext_vector_type(8)));

#define DIM 128

// ---------------- utility: zero a float buffer (grid-stride) ----------------
__global__ void gcn_zero_f32(float* __restrict__ p, int n) {
    int i = blockIdx.x * blockDim.x + threadIdx.x;
    int stride = gridDim.x * blockDim.x;
    for (; i < n; i += stride) p[i] = 0.0f;
}

// ---------------- degree accumulation over edges ----------------
__global__ void gcn_degree(const int* __restrict__ src, const int* __restrict__ dst,
                           float* __restrict__ deg_s, float* __restrict__ deg_d, int ne) {
    int i = blockIdx.x * blockDim.x + threadIdx.x;
    if (i < ne) {
        atomicAdd(&deg_s[src[i]], 1.0f);
        atomicAdd(&deg_d[dst[i]], 1.0f);
    }
}

// ---------------- deg -> deg^{-1/2} with clamp-to-1 (DGL norm='both') ----------------
__global__ void gcn_norm(float* __restrict__ p, int n) {
    int i = blockIdx.x * blockDim.x + threadIdx.x;
    if (i < n) p[i] = rsqrtf(fmaxf(p[i], 1.0f));
}

// ---------------- edge scatter: agg[dst] += h[src] * norm_s[src] ----------------
// one wave32 per edge, 4 contiguous dims per lane (float4 gather, 4 f32 atomics)
__global__ void gcn_scatter(const float* __restrict__ h, const float* __restrict__ norm_s,
                            const int* __restrict__ src, const int* __restrict__ dst,
                            float* __restrict__ agg, int ne) {
    int tid  = blockIdx.x * blockDim.x + threadIdx.x;
    int e    = tid >> 5;
    if (e >= ne) return;
    int lane = tid & 31;
    int s = src[e];
    int d = dst[e];
    float ns = norm_s[s];
    const float4 v = *reinterpret_cast<const float4*>(h + (size_t)s * DIM + lane * 4);
    float* ap = agg + (size_t)d * DIM + lane * 4;
    atomicAdd(ap + 0, v.x * ns);
    atomicAdd(ap + 1, v.y * ns);
    atomicAdd(ap + 2, v.z * ns);
    atomicAdd(ap + 3, v.w * ns);
}

// ---------------- dense layer: out = (agg * norm_d) @ W + b [, ReLU] ----------------
// Block = 256 threads = 8 waves. Block owns 16 output rows; wave w owns columns
// [16w, 16w+16). K=128 walked in steps of 4 with V_WMMA_F32_16X16X4_F32 (exact f32).
//
// Fragment layouts per CDNA5 ISA 7.12.2 (wave32):
//   A (16x4 f32):  lane L holds row M=L%16; VGPR v holds K = 2*(L/16) + v
//   B (4x16 f32):  VGPR v, lane L holds K = v + 2*(L/16), N = L%16
//   C/D (16x16):   VGPR r, lane L holds M = r + 8*(L/16), N = L%16
#define LDA 129  // LDS pitch: odd => lanes 0..15 hit distinct banks on A reads

__global__ void gcn_gemm_wmma(const float* __restrict__ agg, const float* __restrict__ norm_d,
                              const float* __restrict__ W, const float* __restrict__ bias,
                              float* __restrict__ out, int relu) {
    __shared__ float lA[16 * LDA];

    const int m0 = blockIdx.x * 16;
    const int t  = threadIdx.x;

    // cooperative, coalesced load of the 16x128 A-tile, pre-scaled by norm_d
    for (int i = t; i < 16 * DIM; i += 256) {
        int r = i >> 7;       // tile row 0..15
        int k = i & (DIM - 1);
        int row = m0 + r;
        lA[r * LDA + k] = agg[(size_t)row * DIM + k] * norm_d[row];
    }
    __syncthreads();

    const int wave = t >> 5;       // 0..7 -> N-tile
    const int lane = t & 31;
    const int n0   = wave * 16;
    const int half = lane >> 4;    // 0 or 1
    const int l16  = lane & 15;

    const float* Ar = lA + l16 * LDA;       // A row for this lane
    const float* Wc = W + n0 + l16;         // B column for this lane

    v8f c = {};
#pragma unroll
    for (int kk = 0; kk < DIM; kk += 4) {
        const int kbase = kk + half * 2;
        v2f a, b;
        a.x = Ar[kbase + 0];
        a.y = Ar[kbase + 1];
        b.x = Wc[(kbase + 0) * DIM];
        b.y = Wc[(kbase + 1) * DIM];
        c = __builtin_amdgcn_wmma_f32_16x16x4_f32(
                /*neg_a=*/false, a, /*neg_b=*/false, b,
                /*c_mod=*/(short)0, c, /*reuse_a=*/false, /*reuse_b=*/false);
    }

    const float bn = bias[n0 + l16];
#pragma unroll
    for (int r = 0; r < 8; ++r) {
        int row = m0 + r + half * 8;
        float v = c[r] + bn;
        if (relu) v = fmaxf(v, 0.0f);
        out[(size_t)row * DIM + n0 + l16] = v;
    }
}

// ---------------- driver ----------------
extern "C" void kernel_launch(void* const* d_in, const int* in_sizes, int n_in,
                              void* d_out, int out_size, void* d_ws, size_t ws_size,
                              hipStream_t stream) {
    const float* feat = (const float*)d_in[0];
    const float* W1   = (const float*)d_in[1];
    const float* b1   = (const float*)d_in[2];
    const float* W2   = (const float*)d_in[3];
    const float* b2   = (const float*)d_in[4];
    const int*   src  = (const int*)d_in[5];
    const int*   dst  = (const int*)d_in[6];

    const int N = in_sizes[0] / DIM;   // 100000
    const int E = in_sizes[5];         // 1600000

    float* ws     = (float*)d_ws;
    float* norm_s = ws;                // N floats
    float* norm_d = ws + N;            // N floats (contiguous with norm_s)
    float* agg    = ws + 2 * (size_t)N;// N*DIM floats

    float* out = (float*)d_out;

    const int ZB = 256;
    const int aggN = N * DIM;          // 12.8M, fits int

    // ---- degrees -> norms ----
    gcn_zero_f32<<<(2 * N + ZB - 1) / ZB, ZB, 0, stream>>>(norm_s, 2 * N);
    gcn_degree<<<(E + ZB - 1) / ZB, ZB, 0, stream>>>(src, dst, norm_s, norm_d, E);
    gcn_norm<<<(2 * N + ZB - 1) / ZB, ZB, 0, stream>>>(norm_s, 2 * N);

    const int scatterThreads = E * 32; // 51.2M, fits int
    const int scatterBlocks  = (scatterThreads + ZB - 1) / ZB;

    // ---- layer 1: agg = S^T (feat * norm_s); out = relu((agg*norm_d) @ W1 + b1) ----
    gcn_zero_f32<<<(aggN + ZB - 1) / ZB, ZB, 0, stream>>>(agg, aggN);
    gcn_scatter<<<scatterBlocks, ZB, 0, stream>>>(feat, norm_s, src, dst, agg, E);
    gcn_gemm_wmma<<<N / 16, 256, 0, stream>>>(agg, norm_d, W1, b1, out, /*relu=*/1);

    // ---- layer 2: same, reading h1 from d_out, final write to d_out ----
    gcn_zero_f32<<<(aggN + ZB - 1) / ZB, ZB, 0, stream>>>(agg, aggN);
    gcn_scatter<<<scatterBlocks, ZB, 0, stream>>>(out, norm_s, src, dst, agg, E);
    gcn_gemm_wmma<<<N / 16, 256, 0, stream>>>(agg, norm_d, W2, b2, out, /*relu=*/0);
}